// main_net_20486994002610
// MI455X (gfx1250) — compile-verified
//
#include <hip/hip_runtime.h>
#include <hip/hip_bf16.h>
#include <math.h>

#define BB 2
#define NN 4096
#define MM 12288
#define KSEL 64
#define SHH 32
#define HH 64
#define GKK 8
#define NCC 200

typedef __attribute__((ext_vector_type(2))) float v2f;
typedef __attribute__((ext_vector_type(8))) float v8f;

// ---------------------------------------------------------------------------
// Kernel 1: precompute unit vectors (SoA for coalescing) and |r_pix|^2
// ---------------------------------------------------------------------------
__global__ void prep_kernel(const float* __restrict__ xxx,
                            const float* __restrict__ pix_tp,
                            float* __restrict__ rlos,   // [3][B*N]
                            float* __restrict__ rpix,   // [3][M]
                            float* __restrict__ sqpix)  // [M]
{
    int i = blockIdx.x * blockDim.x + threadIdx.x;
    if (i < BB * NN) {
        int b = i / NN, n = i - b * NN;
        float th = xxx[b * 3 * NN + n];
        float ph = xxx[b * 3 * NN + NN + n];
        float st = sinf(th);
        rlos[0 * BB * NN + i] = st * cosf(ph);
        rlos[1 * BB * NN + i] = st * sinf(ph);
        rlos[2 * BB * NN + i] = cosf(th);
    }
    if (i < MM) {
        float th = pix_tp[i * 2 + 0];
        float ph = pix_tp[i * 2 + 1];
        float st = sinf(th);
        float x = st * cosf(ph), y = st * sinf(ph), z = cosf(th);
        rpix[0 * MM + i] = x;
        rpix[1 * MM + i] = y;
        rpix[2 * MM + i] = z;
        sqpix[i] = x * x + y * y + z * z;
    }
}

// ---------------------------------------------------------------------------
// Kernel 2: per (b,m): cos over N, top-64 selection (register-resident),
// attention MLP + softmax pooling, project to h0[b,m,0:64].
// One 256-thread workgroup per (b,m).
// ---------------------------------------------------------------------------
__global__ void __launch_bounds__(256)
sampler_kernel(const float* __restrict__ xxx,
               const float* __restrict__ rlos,
               const float* __restrict__ rpix,
               const float* __restrict__ aw1, const float* __restrict__ ab1,
               const float* __restrict__ aw2, const float* __restrict__ ab2,
               const float* __restrict__ pw,  const float* __restrict__ pb,
               float* __restrict__ h0)
{
    int bm = blockIdx.x;
    int b = bm / MM, m = bm - b * MM;
    int tid = threadIdx.x;
    int lane = tid & 31, wave = tid >> 5;

    __shared__ float s_rv[8];
    __shared__ int   s_ri[8];
    __shared__ float s_selv[KSEL];
    __shared__ int   s_seli[KSEL];
    __shared__ float s_log[KSEL];
    __shared__ float s_xv[KSEL];
    __shared__ int   s_wini;
    __shared__ float s_pool;

    float px = rpix[m], py = rpix[MM + m], pz = rpix[2 * MM + m];

    const float* rx = rlos + 0 * BB * NN + b * NN;
    const float* ry = rlos + 1 * BB * NN + b * NN;
    const float* rz = rlos + 2 * BB * NN + b * NN;

    // 16 cos values per thread, strided layout: n = s*256 + tid
    float cv[16];
#pragma unroll
    for (int s = 0; s < 16; ++s) {
        int n = s * 256 + tid;
        float c = rx[n] * px + ry[n] * py + rz[n] * pz;
        cv[s] = fminf(1.f, fmaxf(-1.f, c));
    }

    unsigned removed = 0u;
    for (int it = 0; it < KSEL; ++it) {
        float bv = -3.f;
        int   bi = 1 << 30;
#pragma unroll
        for (int s = 0; s < 16; ++s) {
            bool ok = ((removed >> s) & 1u) == 0u;
            int n = s * 256 + tid;
            float v = cv[s];
            if (ok && (v > bv || (v == bv && n < bi))) { bv = v; bi = n; }
        }
#pragma unroll
        for (int off = 16; off > 0; off >>= 1) {
            float ov = __shfl_down(bv, off, 32);
            int   oi = __shfl_down(bi, off, 32);
            if (ov > bv || (ov == bv && oi < bi)) { bv = ov; bi = oi; }
        }
        if (lane == 0) { s_rv[wave] = bv; s_ri[wave] = bi; }
        __syncthreads();
        if (tid == 0) {
            float wv = s_rv[0]; int wi = s_ri[0];
#pragma unroll
            for (int w = 1; w < 8; ++w)
                if (s_rv[w] > wv || (s_rv[w] == wv && s_ri[w] < wi)) { wv = s_rv[w]; wi = s_ri[w]; }
            s_selv[it] = wv; s_seli[it] = wi; s_wini = wi;
        }
        __syncthreads();
        int wi = s_wini;
        if ((wi & 255) == tid) removed |= 1u << (wi >> 8);
    }
    __syncthreads();

    // Attention MLP per selected sample (threads 0..63)
    if (tid < KSEL) {
        int n = s_seli[tid];
        float val = s_selv[tid];
        float xv = xxx[b * 3 * NN + 2 * NN + n];
        float dd = acosf(fminf(1.f, fmaxf(-1.f, val)));
        float logit = ab2[0];
#pragma unroll
        for (int j = 0; j < SHH; ++j) {
            float hj = xv * aw1[j] + dd * aw1[SHH + j] + ab1[j];
            hj = fmaxf(hj, 0.f);
            logit += hj * aw2[j];
        }
        s_log[tid] = logit;
        s_xv[tid]  = xv;
    }
    __syncthreads();

    if (tid == 0) {
        float mx = s_log[0];
        for (int k = 1; k < KSEL; ++k) mx = fmaxf(mx, s_log[k]);
        float sum = 0.f, acc = 0.f;
        for (int k = 0; k < KSEL; ++k) {
            float e = expf(s_log[k] - mx);
            sum += e;
            acc += e * s_xv[k];
        }
        s_pool = acc / sum;
    }
    __syncthreads();

    if (tid < HH) {
        float hv = s_pool * pw[tid] + pb[tid];
        h0[(size_t)bm * HH + tid] = fmaxf(hv, 0.f);
    }
}

// ---------------------------------------------------------------------------
// Kernel 3: kNN graph (top-8 smallest d2, self-excluded). One WG per pixel m.
// ---------------------------------------------------------------------------
__global__ void __launch_bounds__(256)
knn_kernel(const float* __restrict__ rpix,
           const float* __restrict__ sqpix,
           int* __restrict__ nbr)
{
    int m = blockIdx.x;
    int tid = threadIdx.x;
    int lane = tid & 31, wave = tid >> 5;

    __shared__ float s_v[256 * GKK];
    __shared__ int   s_i[256 * GKK];
    __shared__ float s_rv[8];
    __shared__ int   s_ri[8];
    __shared__ int   s_re[8];

    float px = rpix[m], py = rpix[MM + m], pz = rpix[2 * MM + m];
    float sqm = sqpix[m];

    float kv[GKK];
    int   ki[GKK];
#pragma unroll
    for (int s = 0; s < GKK; ++s) { kv[s] = -3.0e38f; ki[s] = -1; }

    for (int s = 0; s < 48; ++s) {
        int j = s * 256 + tid;
        if (j == m) continue;
        float dot = rpix[j] * px + rpix[MM + j] * py + rpix[2 * MM + j] * pz;
        float key = -(sqm + sqpix[j] - 2.f * dot);  // maximize -d2
        if (key > kv[GKK - 1]) {
            kv[GKK - 1] = key; ki[GKK - 1] = j;
#pragma unroll
            for (int t = GKK - 1; t > 0; --t) {
                if (kv[t] > kv[t - 1]) {
                    float tv = kv[t]; kv[t] = kv[t - 1]; kv[t - 1] = tv;
                    int ti = ki[t]; ki[t] = ki[t - 1]; ki[t - 1] = ti;
                }
            }
        }
    }
#pragma unroll
    for (int s = 0; s < GKK; ++s) {
        s_v[tid * GKK + s] = kv[s];
        s_i[tid * GKK + s] = ki[s];
    }
    __syncthreads();

    __shared__ int s_dummy; (void)s_dummy;
    for (int it = 0; it < GKK; ++it) {
        float bv = -3.4e38f; int bi = 1 << 30; int be = -1;
#pragma unroll
        for (int s = 0; s < GKK; ++s) {
            int e = tid * GKK + s;
            float v = s_v[e]; int i2 = s_i[e];
            if (v > bv || (v == bv && i2 < bi)) { bv = v; bi = i2; be = e; }
        }
#pragma unroll
        for (int off = 16; off > 0; off >>= 1) {
            float ov = __shfl_down(bv, off, 32);
            int   oi = __shfl_down(bi, off, 32);
            int   oe = __shfl_down(be, off, 32);
            if (ov > bv || (ov == bv && oi < bi)) { bv = ov; bi = oi; be = oe; }
        }
        if (lane == 0) { s_rv[wave] = bv; s_ri[wave] = bi; s_re[wave] = be; }
        __syncthreads();
        if (tid == 0) {
            float wv = s_rv[0]; int wi = s_ri[0], we = s_re[0];
#pragma unroll
            for (int w = 1; w < 8; ++w)
                if (s_rv[w] > wv || (s_rv[w] == wv && s_ri[w] < wi)) {
                    wv = s_rv[w]; wi = s_ri[w]; we = s_re[w];
                }
            nbr[m * GKK + it] = wi;
            s_v[we] = -3.4e38f;
        }
        __syncthreads();
    }
}

// ---------------------------------------------------------------------------
// Kernel 4: one GNN layer: h_out = relu(agg @ W_rel + b + h_in @ W_root)
// 256 threads = 8 waves; each wave computes a 16x64 output tile via
// V_WMMA_F32_16X16X4_F32 (dual-accumulate: agg@Wrel then h@Wroot).
// Weights staged in LDS (pitch 65 to avoid bank conflicts).
// ---------------------------------------------------------------------------
__global__ void __launch_bounds__(256)
gnn_layer_kernel(const float* __restrict__ h_in, float* __restrict__ h_out,
                 const float* __restrict__ wrel, const float* __restrict__ wroot,
                 const float* __restrict__ bias, const int* __restrict__ nbr)
{
    __shared__ float s_wr[64 * 65];
    __shared__ float s_wt[64 * 65];
    __shared__ float s_b[64];

    int tid = threadIdx.x;
    __builtin_prefetch(wrel, 0, 3);
    __builtin_prefetch(wroot, 0, 3);
    for (int i = tid; i < 64 * 64; i += 256) {
        int r = i >> 6, c = i & 63;
        s_wr[r * 65 + c] = wrel[i];
        s_wt[r * 65 + c] = wroot[i];
    }
    if (tid < 64) s_b[tid] = bias[tid];
    __syncthreads();

    int wave = tid >> 5, lane = tid & 31;
    int rl = lane & 15, half = lane >> 4;
    int tr = blockIdx.x * 128 + wave * 16;     // tile row base (flattened b*M+m)
    int rowA = tr + rl;                        // this lane's A-matrix row
    int b = rowA / MM;                         // tiles never straddle batches (M%128==0)
    int mloc = rowA - b * MM;
    const float* hb = h_in + (size_t)b * MM * HH;
    const float* hrow = h_in + (size_t)rowA * HH;

    int nb[GKK];
#pragma unroll
    for (int t = 0; t < GKK; ++t) nb[t] = nbr[mloc * GKK + t];

    v8f acc[4];
#pragma unroll
    for (int ct = 0; ct < 4; ++ct) {
        float bc = s_b[ct * 16 + rl];
#pragma unroll
        for (int v = 0; v < 8; ++v) acc[ct][v] = bc;
    }

    for (int k = 0; k < 16; ++k) {
        int kb = 4 * k + 2 * half;  // this lane-half's K offset

        // A fragment from h (root path): h[row][kb..kb+1]
        float2 ahp = *(const float2*)(hrow + kb);
        v2f ah; ah.x = ahp.x; ah.y = ahp.y;

        // A fragment from agg: sum of 8 neighbor rows (L2-resident gathers)
        float agx = 0.f, agy = 0.f;
#pragma unroll
        for (int t = 0; t < GKK; ++t) {
            float2 hv = *(const float2*)(hb + (size_t)nb[t] * HH + kb);
            agx += hv.x; agy += hv.y;
        }
        v2f ag; ag.x = agx; ag.y = agy;

#pragma unroll
        for (int ct = 0; ct < 4; ++ct) {
            int c = ct * 16 + rl;
            v2f br; br.x = s_wr[kb * 65 + c]; br.y = s_wr[(kb + 1) * 65 + c];
            v2f bt; bt.x = s_wt[kb * 65 + c]; bt.y = s_wt[(kb + 1) * 65 + c];
            acc[ct] = __builtin_amdgcn_wmma_f32_16x16x4_f32(
                false, ag, false, br, (short)0, acc[ct], false, false);
            acc[ct] = __builtin_amdgcn_wmma_f32_16x16x4_f32(
                false, ah, false, bt, (short)0, acc[ct], false, false);
        }
    }

    // Epilogue: relu + store (C layout: element (M=v+8*half, N=rl) per VGPR v)
#pragma unroll
    for (int ct = 0; ct < 4; ++ct) {
#pragma unroll
        for (int v = 0; v < 8; ++v) {
            int r = tr + v + 8 * half;
            int c = ct * 16 + rl;
            h_out[(size_t)r * HH + c] = fmaxf(acc[ct][v], 0.f);
        }
    }
}

// ---------------------------------------------------------------------------
// Kernel 5: mean-pool over M, then 64->64 relu -> 200 head. One WG per batch.
// ---------------------------------------------------------------------------
__global__ void __launch_bounds__(256)
finalize_kernel(const float* __restrict__ h,
                const float* __restrict__ w1, const float* __restrict__ b1,
                const float* __restrict__ w2, const float* __restrict__ b2,
                float* __restrict__ out)
{
    int b = blockIdx.x, tid = threadIdx.x;
    __shared__ float s_part[256];
    __shared__ float s_gf[64];
    __shared__ float s_hid[64];

    int col = tid & 63, seg = tid >> 6;
    float acc = 0.f;
    for (int m = seg; m < MM; m += 4)
        acc += h[((size_t)b * MM + m) * HH + col];
    s_part[tid] = acc;
    __syncthreads();
    if (tid < 64)
        s_gf[tid] = (s_part[tid] + s_part[tid + 64] + s_part[tid + 128] + s_part[tid + 192]) *
                    (1.0f / (float)MM);
    __syncthreads();
    if (tid < 64) {
        float s = b1[tid];
        for (int c = 0; c < 64; ++c) s += s_gf[c] * w1[c * 64 + tid];
        s_hid[tid] = fmaxf(s, 0.f);
    }
    __syncthreads();
    if (tid < NCC) {
        float s = b2[tid];
        for (int j = 0; j < 64; ++j) s += s_hid[j] * w2[j * NCC + tid];
        out[b * NCC + tid] = s;
    }
}

// ---------------------------------------------------------------------------
extern "C" void kernel_launch(void* const* d_in, const int* in_sizes, int n_in,
                              void* d_out, int out_size, void* d_ws, size_t ws_size,
                              hipStream_t stream)
{
    const float* xxx  = (const float*)d_in[0];
    const float* pix  = (const float*)d_in[1];
    const float* aw1  = (const float*)d_in[2];
    const float* ab1  = (const float*)d_in[3];
    const float* aw2  = (const float*)d_in[4];
    const float* ab2  = (const float*)d_in[5];
    const float* pw   = (const float*)d_in[6];
    const float* pb   = (const float*)d_in[7];
    const float* relw = (const float*)d_in[8];
    const float* relb = (const float*)d_in[9];
    const float* rootw= (const float*)d_in[10];
    const float* ow1  = (const float*)d_in[11];
    const float* ob1  = (const float*)d_in[12];
    const float* ow2  = (const float*)d_in[13];
    const float* ob2  = (const float*)d_in[14];
    float* out = (float*)d_out;
    (void)in_sizes; (void)n_in; (void)out_size; (void)ws_size;

    char* ws = (char*)d_ws;
    size_t off = 0;
    auto alloc = [&](size_t bytes) -> void* {
        void* p = ws + off;
        off += (bytes + 255) & ~(size_t)255;
        return p;
    };
    float* rlos  = (float*)alloc(sizeof(float) * 3 * BB * NN);
    float* rpix  = (float*)alloc(sizeof(float) * 3 * MM);
    float* sqpix = (float*)alloc(sizeof(float) * MM);
    int*   nbr   = (int*)  alloc(sizeof(int)   * MM * GKK);
    float* ha    = (float*)alloc(sizeof(float) * BB * MM * HH);
    float* hb    = (float*)alloc(sizeof(float) * BB * MM * HH);

    prep_kernel<<<(MM + 255) / 256, 256, 0, stream>>>(xxx, pix, rlos, rpix, sqpix);
    sampler_kernel<<<BB * MM, 256, 0, stream>>>(xxx, rlos, rpix, aw1, ab1, aw2, ab2, pw, pb, ha);
    knn_kernel<<<MM, 256, 0, stream>>>(rpix, sqpix, nbr);

    int gnn_grid = (BB * MM) / 128;
    gnn_layer_kernel<<<gnn_grid, 256, 0, stream>>>(ha, hb, relw + 0 * HH * HH, rootw + 0 * HH * HH, relb + 0 * HH, nbr);
    gnn_layer_kernel<<<gnn_grid, 256, 0, stream>>>(hb, ha, relw + 1 * HH * HH, rootw + 1 * HH * HH, relb + 1 * HH, nbr);
    gnn_layer_kernel<<<gnn_grid, 256, 0, stream>>>(ha, hb, relw + 2 * HH * HH, rootw + 2 * HH * HH, relb + 2 * HH, nbr);

    finalize_kernel<<<BB, 256, 0, stream>>>(hb, ow1, ob1, ow2, ob2, out);
}